// SmoothingModule_773094113546
// MI455X (gfx1250) — compile-verified
//
#include <hip/hip_runtime.h>

typedef __attribute__((ext_vector_type(2))) float v2f;
typedef __attribute__((ext_vector_type(8))) float v8f;

namespace {
constexpr int kB   = 4;
constexpr int kT   = 4096;
constexpr int kD   = 512;
constexpr int kC   = 16;              // chunk = WMMA tile M/N
constexpr int kSeg = 16;              // sequence-parallel segments
constexpr int kSegLen = kT / kSeg;    // 256 rows per segment
constexpr int kNT  = 4;               // 16-col N-tiles per wave -> 64 cols
constexpr int kNW  = kD / (16 * kNT); // d-tile groups = 8
}

// Walk one T-segment in 16-row chunks with the chunked-scan identity
//   z[t] = q[t]*z[t-1] + P[t]*x[t]
// Intra-chunk: triangular M_loc[t,k] = P[k]*exp(L[t]-L[k]) applied via
// v_wmma_f32_16x16x4_f32 (K split 4x4), A fragments + row decays shared
// across kNT N-tiles. Returns segment total log-decay (sum log q).
template<bool STORE>
__device__ __forceinline__ float run_segment(
    const float* __restrict__ zb,   // z_hat[b]
    const float* __restrict__ Pb,   // P[b]
    float*       __restrict__ ob,   // out[b]  (used when STORE)
    int t0, int dbase, int len, float h[kNT])
{
  const int lane = threadIdx.x & 31;
  const int l16  = lane & 15;       // col / row-within-half index
  const int hf   = lane >> 4;       // 0: lanes 0-15, 1: lanes 16-31
  float Ltot = 0.0f;

  for (int c0 = t0; c0 < t0 + kSegLen; c0 += kC) {
    // ---- chunk gate math, replicated identically in both halves ----
    float Praw = Pb[c0 + l16];
    float Pc   = fminf(fmaxf(Praw, 0.0f), 1.0f - 1e-6f);
    float Pk   = fmaxf(Pc, 1e-6f);          // exp(log P) after clip
    float L    = __logf(1.0f - Pc);         // log q[t], t = c0 + l16
    #pragma unroll
    for (int off = 1; off < 16; off <<= 1) {  // inclusive 16-lane scan
      float up = __shfl_up(L, off, 16);
      L += (l16 >= off) ? up : 0.0f;
    }
    const float Lt = L;
    const float E  = __expf(L);             // decay chunk-entry -> row t
    Ltot += __shfl(L, 15, 16);

    // prefetch next chunk's 16 rows x 64 cols (two 128B lines per row)
    const int pr = (c0 + kC < kT) ? (c0 + kC) : c0;
    __builtin_prefetch(&zb[(size_t)(pr + l16) * kD + dbase + hf * 32], 0, 1);

    // ---- A fragments (ISA 16x4 f32 layout), shared across N-tiles ----
    v2f a[4];
    #pragma unroll
    for (int g = 0; g < 4; ++g) {
      const int k0 = 4 * g + 2 * hf, k1 = k0 + 1;
      const float Lk0 = __shfl(L,  k0, 16), Lk1 = __shfl(L,  k1, 16);
      const float Pk0 = __shfl(Pk, k0, 16), Pk1 = __shfl(Pk, k1, 16);
      a[g].x = (k0 <= l16) ? Pk0 * __expf(Lt - Lk0) : 0.0f;
      a[g].y = (k1 <= l16) ? Pk1 * __expf(Lt - Lk1) : 0.0f;
    }
    // per-row decays for carry folding (C/D layout rows i, i+8)
    float dec[8];
    #pragma unroll
    for (int i = 0; i < 8; ++i) dec[i] = __shfl(E, i + 8 * hf, 16);

    // ---- kNT N-tiles: 16 WMMAs per chunk against shared A ----
    #pragma unroll
    for (int nt = 0; nt < kNT; ++nt) {
      const int dcol = dbase + nt * 16 + l16;
      v8f acc = {};
      #pragma unroll
      for (int g = 0; g < 4; ++g) {
        const int k0 = 4 * g + 2 * hf;
        v2f bf;
        bf.x = zb[(size_t)(c0 + k0    ) * kD + dcol];
        bf.y = zb[(size_t)(c0 + k0 + 1) * kD + dcol];
        acc = __builtin_amdgcn_wmma_f32_16x16x4_f32(
            false, a[g], false, bf, (short)0, acc, false, false);
      }
      #pragma unroll
      for (int i = 0; i < 8; ++i) {
        acc[i] += dec[i] * h[nt];
        if (STORE) {
          const int row = c0 + i + 8 * hf;
          ob[(size_t)row * kD + dcol] = (row < len) ? acc[i] : 0.0f;
        }
      }
      // row 15 of chunk = acc[7] in lanes 16-31 (M=15, N=lane-16)
      h[nt] = __shfl(acc[7], 16 + l16, 32);
    }
  }
  return Ltot;
}

// Pass 1: per-segment boundary state with zero carry-in + total log-decay.
__global__ __launch_bounds__(32) void seg_state_kernel(
    const float* __restrict__ z_hat, const float* __restrict__ P,
    float* __restrict__ hloc, float* __restrict__ Lseg)
{
  const int id   = blockIdx.x;               // b*kSeg*kNW + seg*kNW + dgrp
  const int dgrp = id % kNW;
  const int seg  = (id / kNW) % kSeg;
  const int b    = id / (kNW * kSeg);
  const int dbase = dgrp * (16 * kNT);
  const int lane = threadIdx.x & 31;
  const int l16 = lane & 15, hf = lane >> 4;

  float h[kNT] = {0.f, 0.f, 0.f, 0.f};
  const float Ltot = run_segment<false>(
      z_hat + (size_t)b * kT * kD, P + (size_t)b * kT,
      nullptr, seg * kSegLen, dbase, 0, h);

  if (hf == 0) {
    #pragma unroll
    for (int nt = 0; nt < kNT; ++nt)
      hloc[(size_t)(b * kSeg + seg) * kD + dbase + nt * 16 + l16] = h[nt];
  }
  if (lane == 0 && dgrp == 0) Lseg[b * kSeg + seg] = Ltot;
}

// Pass 2: tiny sequential combine of segment carries (per (b,d) column).
__global__ __launch_bounds__(256) void seg_combine_kernel(
    const float* __restrict__ hloc, const float* __restrict__ Lseg,
    float* __restrict__ hin)
{
  const int gid = blockIdx.x * blockDim.x + threadIdx.x;  // 0 .. B*D-1
  const int b = gid / kD;
  const int d = gid % kD;
  float h = 0.0f;
  #pragma unroll
  for (int s = 0; s < kSeg; ++s) {
    hin[(size_t)(b * kSeg + s) * kD + d] = h;
    h = __expf(Lseg[b * kSeg + s]) * h + hloc[(size_t)(b * kSeg + s) * kD + d];
  }
}

// Pass 3: stored scan per segment starting from combined carry.
__global__ __launch_bounds__(32) void seg_scan_kernel(
    const float* __restrict__ z_hat, const float* __restrict__ P,
    const int* __restrict__ lengths, const float* __restrict__ hin,
    float* __restrict__ out)
{
  const int id   = blockIdx.x;
  const int dgrp = id % kNW;
  const int seg  = (id / kNW) % kSeg;
  const int b    = id / (kNW * kSeg);
  const int dbase = dgrp * (16 * kNT);
  const int lane = threadIdx.x & 31;
  const int l16 = lane & 15;
  const int len = lengths[b];

  float h[kNT];
  #pragma unroll
  for (int nt = 0; nt < kNT; ++nt)
    h[nt] = hin[(size_t)(b * kSeg + seg) * kD + dbase + nt * 16 + l16];

  run_segment<true>(z_hat + (size_t)b * kT * kD, P + (size_t)b * kT,
                    out + (size_t)b * kT * kD,
                    seg * kSegLen, dbase, len, h);
}

extern "C" void kernel_launch(void* const* d_in, const int* in_sizes, int n_in,
                              void* d_out, int out_size, void* d_ws, size_t ws_size,
                              hipStream_t stream) {
  const float* z_hat   = (const float*)d_in[0];
  const float* P       = (const float*)d_in[1];
  const int*   lengths = (const int*)d_in[2];
  float*       out     = (float*)d_out;

  (void)in_sizes; (void)n_in; (void)out_size; (void)ws_size;

  float* hloc = (float*)d_ws;                  // [B][Seg][D]
  float* hin  = hloc + (size_t)kB * kSeg * kD; // [B][Seg][D]
  float* Lseg = hin  + (size_t)kB * kSeg * kD; // [B][Seg]

  const int heavy_grid = kB * kSeg * kNW;      // 512 independent waves
  seg_state_kernel<<<heavy_grid, 32, 0, stream>>>(z_hat, P, hloc, Lseg);
  seg_combine_kernel<<<(kB * kD) / 256, 256, 0, stream>>>(hloc, Lseg, hin);
  seg_scan_kernel<<<heavy_grid, 32, 0, stream>>>(z_hat, P, lengths, hin, out);
}